// Transformer_81535659147299
// MI455X (gfx1250) — compile-verified
//
#include <hip/hip_runtime.h>
#include <hip/hip_bf16.h>

typedef __attribute__((ext_vector_type(2))) float v2f;
typedef __attribute__((ext_vector_type(8))) float v8f;

#if __has_builtin(__builtin_amdgcn_wmma_f32_16x16x4_f32)
#define HAS_WMMA_F32X4 1
#else
#define HAS_WMMA_F32X4 0
#endif

#define K_SRF   3.0f
#define K_SRF2  9.0f
#define K_LN8   2.0794415416798357f   /* log(GAMMA/ALPHA) = log(8) */

__device__ __forceinline__ int iabs_(int v) { return v < 0 ? -v : v; }

// Matches reference _bucket_base (RES=0.5 -> x*2, ALPHA=2, BETA=8, GAMMA=16).
// jnp.round is round-half-even -> rintf (default RNE on device).
// Only executed under the (rare) mask, so the faithful f32 division is kept.
__device__ __forceinline__ float bucket_base(float d) {
    float x  = d * 2.0f;                                // d / RES
    float xa = fabsf(x);
    float lr = logf(fmaxf(xa, 1e-6f) * 0.5f) / K_LN8;   // /ALPHA exact (pow2), /log(8)
    float sup = fminf(rintf(2.0f - 6.0f * lr), 8.0f);   // ALPHA-(BETA-ALPHA)*lr, clamp BETA
    float sgn = (x > 0.0f) ? 1.0f : ((x < 0.0f) ? -1.0f : 0.0f);
    sup = sgn * sup;
    return 8.0f + ((xa <= 2.0f) ? rintf(x) : sup);      // BETA + ...
}

// One wave (32 lanes) owns a 16x16 (i,j) pair tile. 8 waves/block -> 32(i) x 64(j).
// WMMA f32 16x16x4 computes the Gram tile g[m][n] = dot(xyz[i0+m], xyz[j0+n]),
// K=3 zero-padded to 4; r^2 = |xi|^2 + |xj|^2 - 2g feeds the radius cutoff.
// All per-i-row derived data is staged once per block in LDS (32 rows x 48 B).
__global__ __launch_bounds__(256) void pairs_kernel(const float* __restrict__ xyz,
                                                    const int*   __restrict__ gidx,
                                                    float* __restrict__ out, int N) {
    __shared__ float4 sPos[32];     // {x, y, z, |x|^2}
    __shared__ int4   sMetaA[32];   // {batch, block, c0, c1}
    __shared__ int4   sMetaB[32];   // {c2, ceil(x/3), ceil(y/3), block>>1}

    const int lane = threadIdx.x & 31;
    const int wave = threadIdx.x >> 5;
    const int hi   = lane >> 4;        // lane half (WMMA M/K split)
    const int ln   = lane & 15;
    const int iBlk = blockIdx.y * 32;                 // block's 32-row i range
    const int i0   = iBlk + (wave >> 2) * 16;         // this wave's 16-row i tile
    const int j0   = blockIdx.x * 64 + (wave & 3) * 16;

    // ---- stage i-side row data into LDS (wave 0); exact reference /3.0f division
    //      now runs 2x per row per block instead of 2x per pair.
    if (threadIdx.x < 32) {
        const int gi = min(iBlk + (int)threadIdx.x, N - 1);
        const float x = xyz[gi * 3 + 0];
        const float y = xyz[gi * 3 + 1];
        const float z = xyz[gi * 3 + 2];
        sPos[threadIdx.x] = make_float4(x, y, z, x * x + y * y + z * z);
        const int ba = gidx[gi * 5 + 0], bl = gidx[gi * 5 + 1];
        sMetaA[threadIdx.x] = make_int4(ba, bl, gidx[gi * 5 + 2], gidx[gi * 5 + 3]);
        sMetaB[threadIdx.x] = make_int4(gidx[gi * 5 + 4],
                                        (int)ceilf(x / K_SRF),
                                        (int)ceilf(y / K_SRF),
                                        bl >> 1);
    }

    // ---- WMMA operands (independent of LDS): 32-bit 16x4 A layout (ISA 7.12.2):
    //   VGPR0: lanes 0-15 -> K=0, lanes 16-31 -> K=2
    //   VGPR1: lanes 0-15 -> K=1, lanes 16-31 -> K=3 (zero pad)
    // B (4x16) mirrored: same K-slot placement, N = lane index.
    const int ia = min(i0 + ln, N - 1);
    const int ja = min(j0 + ln, N - 1);
    v2f a, b;
    a.x = xyz[ia * 3 + (hi ? 2 : 0)];
    a.y = hi ? 0.0f : xyz[ia * 3 + 1];
    b.x = xyz[ja * 3 + (hi ? 2 : 0)];
    b.y = hi ? 0.0f : xyz[ja * 3 + 1];
    v8f gram = {};
#if HAS_WMMA_F32X4
    // 8 args: (neg_a, A, neg_b, B, c_mod, C, reuse_a, reuse_b)
    gram = __builtin_amdgcn_wmma_f32_16x16x4_f32(false, a, false, b, (short)0,
                                                 gram, false, false);
#endif

    // ---- per-lane j-side data (fixed for this lane across the 8 C rows)
    const int   j   = j0 + ln;
    const int   jc  = min(j, N - 1);
    const float xj  = xyz[jc * 3 + 0];
    const float yj  = xyz[jc * 3 + 1];
    const float zj  = xyz[jc * 3 + 2];
    const int   batch_j = gidx[jc * 5 + 0];
    const int   block_j = gidx[jc * 5 + 1];
    const int   cj0 = gidx[jc * 5 + 2], cj1 = gidx[jc * 5 + 3], cj2 = gidx[jc * 5 + 4];
    const float nj  = xj * xj + yj * yj + zj * zj;
    const int   cxj = (int)ceilf(xj / K_SRF);
    const int   cyj = (int)ceilf(yj / K_SRF);
    const int   scan_j = block_j >> 1;     // block in [0,8): // SLICES_PER_ROT
    const bool  jok = (j < N);

    __syncthreads();

    const size_t NN  = (size_t)N * (size_t)N;
    float* __restrict__ outB = out + 3 * NN;   // buckets
    float* __restrict__ outS = out + 4 * NN;   // dscanid
    float* __restrict__ outM = out + 5 * NN;   // mask

    #pragma unroll
    for (int v = 0; v < 8; ++v) {
        const int widx = (wave >> 2) * 16 + hi * 8 + v;  // row within block tile
        const int i    = iBlk + widx;                    // C/D row: M = v + 8*laneHalf
        const float4 P  = sPos[widx];                    // ds_load_b128 broadcast
        const int4   MA = sMetaA[widx];
        const int4   MB = sMetaB[widx];

        const float dx = P.x - xj, dy = P.y - yj, dz = P.z - zj;

#if HAS_WMMA_F32X4
        const float r2 = fmaf(-2.0f, gram[v], P.w + nj);
#else
        const float r2 = dx * dx + dy * dy + dz * dz;
#endif
        const bool keepr    = r2 <= K_SRF2;
        const bool batch_eq = (MA.x == batch_j);
        const bool block_le = (MA.y <= block_j);
        const bool forcekeep = (MA.y == block_j) &&
                               (iabs_(MA.z - cj0) <= 1) &&
                               (iabs_(MA.w - cj1) <= 1) &&
                               (iabs_(MB.x - cj2) <= 1);
        const bool keep_coarse = (iabs_(MB.y - cxj) <= 1) && (iabs_(MB.z - cyj) <= 1);
        const bool mask = batch_eq && block_le &&
                          (forcekeep || keep_coarse) && (forcekeep || keepr);

        const float dxm = mask ? dx : 0.0f;
        const float dym = mask ? dy : 0.0f;
        const float dzm = mask ? dz : 0.0f;

        float bf = 0.0f, dsf = 0.0f;
        if (mask) {
            // bucket values are exact small integers in f32 (<= 4912 < 2^24);
            // int cast + float reconvert is an identity, so write the float sum.
            bf  = 289.0f * bucket_base(dx) + 17.0f * bucket_base(dy) + bucket_base(dz);
            dsf = (float)(scan_j - MB.w);
        }

        if (jok && i < N) {
            const size_t p = (size_t)i * (size_t)N + (size_t)j;
            // Streaming 226 MB > 192 MB L2: non-temporal stores.
            __builtin_nontemporal_store(dxm, out + 3 * p + 0);
            __builtin_nontemporal_store(dym, out + 3 * p + 1);
            __builtin_nontemporal_store(dzm, out + 3 * p + 2);
            __builtin_nontemporal_store(bf,  outB + p);
            __builtin_nontemporal_store(dsf, outS + p);
            __builtin_nontemporal_store(mask ? 1.0f : 0.0f, outM + p);
        }
    }
}

extern "C" void kernel_launch(void* const* d_in, const int* in_sizes, int n_in,
                              void* d_out, int out_size, void* d_ws, size_t ws_size,
                              hipStream_t stream) {
    const float* xyz  = (const float*)d_in[0];   // (N,3) f32
    const int*   grid = (const int*)d_in[1];     // (N,5) i32
    float*       out  = (float*)d_out;           // dxyz_m | buckets | dscanid | mask
    const int N = in_sizes[0] / 3;

    dim3 block(256);                              // 8 wave32 -> 32x64 pair tile
    dim3 grd((unsigned)((N + 63) / 64), (unsigned)((N + 31) / 32));
    hipLaunchKernelGGL(pairs_kernel, grd, block, 0, stream, xyz, grid, out, N);
}